// ScaledDotProductAttention_2018634629272
// MI455X (gfx1250) — compile-verified
//
#include <hip/hip_runtime.h>

typedef __attribute__((ext_vector_type(2))) float v2f;
typedef __attribute__((ext_vector_type(4))) float v4f;
typedef __attribute__((ext_vector_type(8))) float v8f;

#define B_N   8
#define L_N   4096
#define D_N   64
#define MHALF 128
#define WIN   257            // 2*MHALF+1
#define QT    16             // query rows per wave
#define NCT   18             // 16-wide column tiles covering the band union
#define SPAN  (NCT * 16)     // 288 >= 272 (max band union for a 16-row tile)
#define SCALE 0.125f         // 1/sqrt(64)

// ---------------------------------------------------------------------------
// Kernel 1: stream zeros over the dense attention output (outside-window = 0).
// 536 MB >> 192 MB L2, so use non-temporal stores.
// ---------------------------------------------------------------------------
__global__ __launch_bounds__(256) void attn_zero_kernel(v4f* __restrict__ attn4,
                                                        long long n4) {
  long long i = (long long)blockIdx.x * blockDim.x + threadIdx.x;
  long long stride = (long long)gridDim.x * blockDim.x;
  v4f z = {0.0f, 0.0f, 0.0f, 0.0f};
  for (; i < n4; i += stride) {
    __builtin_nontemporal_store(z, attn4 + i);
  }
}

// ---------------------------------------------------------------------------
// Kernel 2: one wave32 per (batch, 16-row query tile).
//   QK^T band via v_wmma_f32_16x16x4_f32 -> LDS -> softmax -> NT store band
//   -> P*V via v_wmma_f32_16x16x4_f32 (A-fragments from LDS).
// ---------------------------------------------------------------------------
__global__ __launch_bounds__(32) void swa_band_kernel(const float* __restrict__ q,
                                                      const float* __restrict__ k,
                                                      const float* __restrict__ v,
                                                      float* __restrict__ out,
                                                      float* __restrict__ attn) {
  __shared__ float sP[QT][SPAN];   // 18 KB: scores, then probabilities

  const int lane  = threadIdx.x;
  const int laneM = lane & 15;     // N / M-row index within half-wave
  const int hseg  = lane >> 4;     // 0: lanes 0-15, 1: lanes 16-31

  const int tile  = blockIdx.x;
  const int b     = tile / (L_N / QT);
  const int qt    = tile % (L_N / QT);
  const int qbase = qt * QT;

  // window start for the first row of the tile -> 16-aligned band origin
  int s0 = qbase - MHALF;
  if (s0 < 0) s0 = 0;
  if (s0 > L_N - WIN) s0 = L_N - WIN;
  const int c0 = s0 & ~15;

  const float* qb = q + (size_t)b * L_N * D_N;
  const float* kb = k + (size_t)b * L_N * D_N;
  const float* vb = v + (size_t)b * L_N * D_N;

  // ---- preload A fragments of Q: lane holds row (qbase+laneM),
  //      K = i*4 + 2*hseg + {0,1} (ISA 32-bit 16x4 A layout) ----
  v2f qa[16];
  {
    const float* qrow = qb + (size_t)(qbase + laneM) * D_N + 2 * hseg;
    #pragma unroll
    for (int i = 0; i < 16; ++i)
      qa[i] = *(const v2f*)(qrow + 4 * i);
  }

  // ---- QK^T over 18 column tiles ----
  for (int t = 0; t < NCT; ++t) {
    const int col  = c0 + t * 16 + laneM;                 // B fragment N-column
    const int colc = col < L_N ? col : L_N - 1;           // clamp (masked later)
    const float* krow = kb + (size_t)colc * D_N + 2 * hseg;

    v8f acc = {};
    #pragma unroll
    for (int i = 0; i < 16; ++i) {
      v2f bf = *(const v2f*)(krow + 4 * i);               // B[k][n] = K[col][k]
      acc = __builtin_amdgcn_wmma_f32_16x16x4_f32(
          false, qa[i], false, bf, (short)0, acc, false, false);
    }

    // mask + scale, stash into LDS (C layout: row = r + 8*hseg, col = laneM)
    #pragma unroll
    for (int r = 0; r < 8; ++r) {
      const int rowb = r + 8 * hseg;
      const int qr   = qbase + rowb;
      int st = qr - MHALF;
      if (st < 0) st = 0;
      if (st > L_N - WIN) st = L_N - WIN;
      const bool ok = (col >= st) && (col < st + WIN) && (col < L_N);
      sP[rowb][t * 16 + laneM] = ok ? acc[r] * SCALE : -__builtin_inff();
    }
  }
  __syncthreads();

  // ---- softmax across the band: 2 lanes per row (144 cols each) ----
  {
    const int segw = SPAN / 2;                      // 144
    const int base = hseg * segw;
    float m = -__builtin_inff();
    for (int c = 0; c < segw; ++c) m = fmaxf(m, sP[laneM][base + c]);
    m = fmaxf(m, __shfl_xor(m, 16, 32));
    float s = 0.0f;
    for (int c = 0; c < segw; ++c) s += __expf(sP[laneM][base + c] - m);
    s += __shfl_xor(s, 16, 32);
    const float inv = 1.0f / s;
    for (int c = 0; c < segw; ++c) {
      const float x = sP[laneM][base + c];
      sP[laneM][base + c] = __expf(x - m) * inv;    // masked -> exp(-inf)=0
    }
  }
  __syncthreads();

  // ---- stream the probability band to the dense attention matrix ----
  {
    float* ab = attn + ((size_t)b * L_N + qbase) * L_N;
    const int n4 = QT * (SPAN / 4);                 // 16 * 72 = 1152 vec4s
    for (int i = lane; i < n4; i += 32) {
      const int row = i / (SPAN / 4);
      const int c4  = i % (SPAN / 4);
      const int col = c0 + c4 * 4;                  // 4-aligned; L%4==0
      if (col < L_N) {
        v4f p = *(const v4f*)&sP[row][c4 * 4];
        __builtin_nontemporal_store(p, (v4f*)(ab + (size_t)row * L_N + col));
      }
    }
  }

  // ---- P*V: out tile 16x64, K dim = band columns, 4 N-tiles of 16 ----
  v8f oacc[4] = {};
  for (int kc = 0; kc < SPAN; kc += 4) {
    // A fragment from LDS: row = laneM, K = kc + 2*hseg + {0,1}
    v2f a = *(const v2f*)&sP[laneM][kc + 2 * hseg];
    const int c  = c0 + kc + 2 * hseg;
    const int ca = c < L_N ? c : L_N - 1;           // clamp; p==0 masks garbage
    const int cb = (c + 1) < L_N ? (c + 1) : L_N - 1;
    #pragma unroll
    for (int nt = 0; nt < 4; ++nt) {
      const int od = nt * 16 + laneM;               // B fragment N = out dim
      v2f bf;
      bf.x = vb[(size_t)ca * D_N + od];
      bf.y = vb[(size_t)cb * D_N + od];
      oacc[nt] = __builtin_amdgcn_wmma_f32_16x16x4_f32(
          false, a, false, bf, (short)0, oacc[nt], false, false);
    }
  }

  // ---- store out (C layout: row = r + 8*hseg, col = nt*16 + laneM) ----
  float* ob = out + ((size_t)b * L_N + qbase) * D_N;
  #pragma unroll
  for (int nt = 0; nt < 4; ++nt) {
    #pragma unroll
    for (int r = 0; r < 8; ++r) {
      const int rowb = r + 8 * hseg;
      ob[(size_t)rowb * D_N + nt * 16 + laneM] = oacc[nt][r];
    }
  }
}

// ---------------------------------------------------------------------------
extern "C" void kernel_launch(void* const* d_in, const int* in_sizes, int n_in,
                              void* d_out, int out_size, void* d_ws, size_t ws_size,
                              hipStream_t stream) {
  const float* q = (const float*)d_in[0];
  const float* k = (const float*)d_in[1];
  const float* v = (const float*)d_in[2];

  float* out  = (float*)d_out;                              // [B, L, D]
  float* attn = out + (size_t)B_N * L_N * D_N;              // [B, L, L]

  // 1) zero the dense attention output (outside-window entries are exactly 0)
  const long long n4 = (long long)B_N * L_N * L_N / 4;
  const int zgrid = (int)((n4 + 255) / 256);
  attn_zero_kernel<<<zgrid, 256, 0, stream>>>((v4f*)attn, n4);

  // 2) band-only attention: one wave per (batch, 16-row query tile)
  const int blocks = B_N * (L_N / QT);                      // 2048
  swa_band_kernel<<<blocks, 32, 0, stream>>>(q, k, v, out, attn);
}